// BaselineGCN_36429912604729
// MI455X (gfx1250) — compile-verified
//
#include <hip/hip_runtime.h>
#include <hip/hip_bf16.h>

typedef __attribute__((ext_vector_type(2))) float v2f;
typedef __attribute__((ext_vector_type(8))) float v8f;

// ---------------------------------------------------------------------------
// Zero-fill (atomic destinations must start at 0 every call)
// ---------------------------------------------------------------------------
__global__ __launch_bounds__(256) void zero_f4(float4* __restrict__ p, int n4) {
    int i = blockIdx.x * 256 + threadIdx.x;
    if (i < n4) p[i] = make_float4(0.f, 0.f, 0.f, 0.f);
}

// ---------------------------------------------------------------------------
// Dense GEMM  Out[M x NCOLS] = (RELU_IN ? relu(A) : A)[M x 128] * W[128 x NCOLS]
// using V_WMMA_F32_16X16X4_F32.  Block = 256 threads = 8 waves = 8 16x16 tiles.
//   NCOLS=128, TM=16 : waves cover 16 rows x 128 cols
//   NCOLS=64,  TM=32 : waves cover 32 rows x 64 cols
// A tile staged in LDS with padded stride (132) to avoid bank conflicts.
// ---------------------------------------------------------------------------
template <int NCOLS, int TM, bool RELU_IN>
__global__ __launch_bounds__(256) void gemm_wmma_f32(const float* __restrict__ A,
                                                     const float* __restrict__ W,
                                                     float* __restrict__ Out,
                                                     int M) {
    constexpr int LDA = 132;                  // padded LDS row stride (floats)
    __shared__ float ldsA[TM * LDA];

    const int tid     = threadIdx.x;
    const int rowBase = blockIdx.x * TM;

    // ---- stage A tile (TM x 128 fp32) into LDS, fusing optional ReLU ----
    constexpr int NV4 = TM * 32;              // float4 elements in the tile
    for (int i = tid; i < NV4; i += 256) {
        const int r  = i >> 5;
        const int c4 = i & 31;
        float4 v;
        if (rowBase + r < M) {
            v = *(const float4*)(A + (size_t)(rowBase + r) * 128 + c4 * 4);
        } else {
            v = make_float4(0.f, 0.f, 0.f, 0.f);
        }
        if (RELU_IN) {
            v.x = fmaxf(v.x, 0.f); v.y = fmaxf(v.y, 0.f);
            v.z = fmaxf(v.z, 0.f); v.w = fmaxf(v.w, 0.f);
        }
        *(float4*)(&ldsA[r * LDA + c4 * 4]) = v;
    }
    __syncthreads();

    // ---- per-wave 16x16 output tile ----
    const int w    = tid >> 5;                // wave id within block (wave32)
    const int lane = tid & 31;
    constexpr int CT = NCOLS / 16;            // column tiles
    const int colBase = (w % CT) * 16;
    const int trow    = (w / CT) * 16;        // row tile base inside LDS tile
    const int half    = lane >> 4;            // 0: K=k0..k0+1, 1: K=k0+2..k0+3
    const int l16     = lane & 15;

    v8f c = {};
    const float* wp = W + colBase + l16;

#pragma unroll
    for (int k0 = 0; k0 < 128; k0 += 4) {
        const int k = k0 + half * 2;
        // A fragment: row = l16 of this tile, two consecutive K values
        v2f a = *(const v2f*)(&ldsA[(trow + l16) * LDA + k]);
        // B fragment: rows k, k+1 of W at column colBase+l16
        v2f b;
        b.x = wp[(size_t)k * NCOLS];
        b.y = wp[(size_t)(k + 1) * NCOLS];
        c = __builtin_amdgcn_wmma_f32_16x16x4_f32(
                /*neg_a=*/false, a, /*neg_b=*/false, b,
                /*c_mod=*/(short)0, c, /*reuse_a=*/false, /*reuse_b=*/false);
    }

    // ---- store C: VGPR v -> row (trow + v + 8*half), col l16 ----
    const int orow = rowBase + trow + half * 8;
    float* op = Out + (size_t)orow * NCOLS + colBase + l16;
#pragma unroll
    for (int v = 0; v < 8; ++v) {
        if (orow + v < M) op[(size_t)v * NCOLS] = c[v];
    }
}

// ---------------------------------------------------------------------------
// Edge scatter:  Out[dst] += w * H[src]   (one wave per edge, F channels)
// F=128 -> float4/lane, F=64 -> float2/lane.  Hardware fp32 atomics hit L2
// (all destinations are L2-resident on MI455X's 192MB L2).
// ---------------------------------------------------------------------------
template <int F>
__global__ __launch_bounds__(256) void scatter_edges(const float* __restrict__ H,
                                                     const long long* __restrict__ ei,
                                                     const float* __restrict__ ew,
                                                     float* __restrict__ Out,
                                                     int nE) {
    const int wid  = (blockIdx.x * 256 + threadIdx.x) >> 5;
    const int lane = threadIdx.x & 31;
    if (wid >= nE) return;

    const long long s = ei[wid];              // edge_index row 0 = src
    const long long d = ei[(size_t)nE + wid]; // edge_index row 1 = dst
    const float     w = ew[wid];

    if (F == 128) {
        float4 v = *(const float4*)(H + (size_t)s * F + lane * 4);
        float* o = Out + (size_t)d * F + lane * 4;
        unsafeAtomicAdd(o + 0, v.x * w);
        unsafeAtomicAdd(o + 1, v.y * w);
        unsafeAtomicAdd(o + 2, v.z * w);
        unsafeAtomicAdd(o + 3, v.w * w);
    } else {
        float2 v = *(const float2*)(H + (size_t)s * F + lane * 2);
        float* o = Out + (size_t)d * F + lane * 2;
        unsafeAtomicAdd(o + 0, v.x * w);
        unsafeAtomicAdd(o + 1, v.y * w);
    }
}

// ---------------------------------------------------------------------------
// Launch: zero -> GEMM1 -> scatter1 -> GEMM2(relu fused) -> scatter2
// ws layout: [ bufH : M*128 f32 (reused for h2) | bufAgg : M*128 f32 ]
// ---------------------------------------------------------------------------
extern "C" void kernel_launch(void* const* d_in, const int* in_sizes, int n_in,
                              void* d_out, int out_size, void* d_ws, size_t ws_size,
                              hipStream_t stream) {
    const float*     x  = (const float*)d_in[0];      // [M,128]
    const float*     W1 = (const float*)d_in[1];      // [128,128]
    const float*     W2 = (const float*)d_in[2];      // [128,64]
    const float*     ew = (const float*)d_in[3];      // [E]
    const long long* ei = (const long long*)d_in[4];  // [2,E] int64

    const int M = in_sizes[0] / 128;
    const int E = in_sizes[3];

    float* bufH   = (float*)d_ws;                     // h1 then h2
    float* bufAgg = bufH + (size_t)M * 128;           // layer-1 aggregation
    float* out    = (float*)d_out;                    // [M,64]

    // zero atomic accumulation targets
    {
        const int nAgg4 = (M * 128) / 4;
        const int nOut4 = out_size / 4;
        zero_f4<<<(nAgg4 + 255) / 256, 256, 0, stream>>>((float4*)bufAgg, nAgg4);
        zero_f4<<<(nOut4 + 255) / 256, 256, 0, stream>>>((float4*)out, nOut4);
    }

    // layer 1: h1 = x @ W1 ; agg1[dst] += w * h1[src]
    gemm_wmma_f32<128, 16, false><<<(M + 15) / 16, 256, 0, stream>>>(x, W1, bufH, M);
    {
        const long long threads = (long long)E * 32;
        scatter_edges<128><<<(int)((threads + 255) / 256), 256, 0, stream>>>(
            bufH, ei, ew, bufAgg, E);
    }

    // layer 2: h2 = relu(agg1) @ W2 ; out[dst] += w * h2[src]
    gemm_wmma_f32<64, 32, true><<<(M + 31) / 32, 256, 0, stream>>>(bufAgg, W2, bufH, M);
    {
        const long long threads = (long long)E * 32;
        scatter_edges<64><<<(int)((threads + 255) / 256), 256, 0, stream>>>(
            bufH, ei, ew, out, E);
    }
}